// LearnedPositionalEncoding_52974126628930
// MI455X (gfx1250) — compile-verified
//
#include <hip/hip_runtime.h>
#include <hip/hip_bf16.h>

typedef __attribute__((ext_vector_type(2))) float v2f;
typedef __attribute__((ext_vector_type(8))) float v8f;

#define B_   8
#define N_   4096
#define D_   64
#define K_   16

// ---------------------------------------------------------------------------
// Kernel 1: exact KNN (top-16 by squared distance, stable tie-break by index),
// using the reference's ||a||^2+||b||^2-2ab expansion. Writes coordinate
// deltas to workspace in [B, K, N, 3] layout (keyed for kernel 2's tiles).
// ---------------------------------------------------------------------------
__global__ void knn_delta_kernel(const float* __restrict__ xyz,
                                 float* __restrict__ delta_ws) {
    __shared__ float sx[256], sy[256], sz[256], ssq[256];

    const int tid = threadIdx.x;
    const int blocksPerBatch = N_ / 256;               // 16
    const int b = blockIdx.x / blocksPerBatch;
    const int n = (blockIdx.x % blocksPerBatch) * 256 + tid;

    const float* q = xyz + ((size_t)b * N_ + n) * 3;
    const float qx = q[0], qy = q[1], qz = q[2];
    const float sqq = qx * qx + qy * qy + qz * qz;

    float bd[K_];
    int   bi[K_];
#pragma unroll
    for (int t = 0; t < K_; ++t) { bd[t] = 3.4e38f; bi[t] = 0; }

    for (int tile = 0; tile < N_ / 256; ++tile) {
        __syncthreads();
        const int cidx = tile * 256 + tid;
        const float* c = xyz + ((size_t)b * N_ + cidx) * 3;
        const float cx = c[0], cy = c[1], cz = c[2];
        sx[tid] = cx; sy[tid] = cy; sz[tid] = cz;
        ssq[tid] = cx * cx + cy * cy + cz * cz;
        __syncthreads();

        const int base = tile * 256;
        for (int i = 0; i < 256; ++i) {
            const float d = sqq + ssq[i]
                          - 2.0f * (qx * sx[i] + qy * sy[i] + qz * sz[i]);
            if (d < bd[K_ - 1]) {
                // stable insert: strictly-less pushes down; equal keys keep
                // earlier (lower-index) entries first, matching stable argsort
                float cd = d; int ci = base + i;
#pragma unroll
                for (int t = 0; t < K_; ++t) {
                    const bool less = cd < bd[t];
                    const float td = bd[t]; const int ti = bi[t];
                    bd[t] = less ? cd : td;
                    bi[t] = less ? ci : ti;
                    cd = less ? td : cd;
                    ci = less ? ti : ci;
                }
            }
        }
    }

    // delta[b, n, k, :] = xyz[b,n] - xyz[b, knn[k]]  stored at [b, k, n, :]
#pragma unroll
    for (int k = 0; k < K_; ++k) {
        const float* p = xyz + ((size_t)b * N_ + bi[k]) * 3;
        float* o = delta_ws + (((size_t)b * K_ + k) * N_ + n) * 3;
        o[0] = qx - p[0];
        o[1] = qy - p[1];
        o[2] = qz - p[2];
    }
}

// ---------------------------------------------------------------------------
// Kernel 2: fused MLP + add, one 16x64 output tile per wave using
// V_WMMA_F32_16X16X4_F32 (full f32 precision, matches reference).
//   GEMM1: delta[16x3 (pad K=4)] @ W1[4x64]  (+b1, ReLU)
//   GEMM2: H[16x64] @ W2[64x64]              (+b2, +x, store)
// H is round-tripped through LDS to convert D-layout -> A-layout.
// ---------------------------------------------------------------------------
#define W2_STRIDE 65
#define H_STRIDE  68

__global__ void mlp_add_kernel(const float* __restrict__ delta_ws,
                               const float* __restrict__ x,
                               const float* __restrict__ W1,
                               const float* __restrict__ b1,
                               const float* __restrict__ W2,
                               const float* __restrict__ b2,
                               float* __restrict__ out) {
    __shared__ float lds_w2[64 * W2_STRIDE];        // 16.25 KB
    __shared__ float lds_h[8 * 16 * H_STRIDE];      // 34 KB (per-wave slabs)

    const int tid    = threadIdx.x;
    const int wave   = tid >> 5;
    const int lane   = tid & 31;
    const int lanelo = lane & 15;
    const int hi     = lane >> 4;                   // 0: K/M group 0, 1: group 1

    // Stage W2 (64x64) into LDS, padded stride
    for (int i = tid; i < 64 * 64; i += 256) {
        lds_w2[(i >> 6) * W2_STRIDE + (i & 63)] = W2[i];
    }

    // Tile decomposition: tile = ((b*K + k) * (N/16)) + nt
    const int tileId = blockIdx.x * 8 + wave;
    const int nt  = tileId & 255;                   // N/16 = 256
    const int bk  = tileId >> 8;                    // b*K + k
    const int n0  = nt * 16;

    // ---- GEMM1: A fragment (16x4, K=3 real + zero pad) ----
    const float* drow = delta_ws + (((size_t)bk) * N_ + n0 + lanelo) * 3;
    const float dx = drow[0], dy = drow[1], dz = drow[2];
    v2f aD;
    aD.x = hi ? dz : dx;
    aD.y = hi ? 0.0f : dy;

    const v8f zeroAcc = {};
    v8f h[4];
#pragma unroll
    for (int j = 0; j < 4; ++j) {
        const int col = j * 16 + lanelo;
        const float w0 = W1[0 * 64 + col];
        const float w1 = W1[1 * 64 + col];
        const float w2 = W1[2 * 64 + col];
        v2f bF;
        bF.x = hi ? w2 : w0;
        bF.y = hi ? 0.0f : w1;
        h[j] = __builtin_amdgcn_wmma_f32_16x16x4_f32(
            false, aD, false, bF, (short)0, zeroAcc, false, false);
    }

    __syncthreads();   // lds_w2 ready (also separates any LDS reuse)

    // Bias + ReLU, then stage H tile to LDS in row-major [16][H_STRIDE]
    float* Hs = lds_h + wave * 16 * H_STRIDE;
#pragma unroll
    for (int j = 0; j < 4; ++j) {
        const float bias1 = b1[j * 16 + lanelo];
#pragma unroll
        for (int v = 0; v < 8; ++v) {
            const float hv = fmaxf(h[j][v] + bias1, 0.0f);
            Hs[(v + 8 * hi) * H_STRIDE + j * 16 + lanelo] = hv;
        }
    }

    __syncthreads();   // uniform across block; guarantees wave's H visible

    // ---- GEMM2: P = H @ W2, K=64 in 16 steps of 4 ----
    v8f p[4];
#pragma unroll
    for (int j = 0; j < 4; ++j) p[j] = zeroAcc;

#pragma unroll
    for (int kk = 0; kk < 16; ++kk) {
        const int cbase = kk * 4 + 2 * hi;
        v2f aH;
        aH.x = Hs[lanelo * H_STRIDE + cbase];
        aH.y = Hs[lanelo * H_STRIDE + cbase + 1];
#pragma unroll
        for (int j = 0; j < 4; ++j) {
            const int col = j * 16 + lanelo;
            v2f bF;
            bF.x = lds_w2[(cbase + 0) * W2_STRIDE + col];
            bF.y = lds_w2[(cbase + 1) * W2_STRIDE + col];
            p[j] = __builtin_amdgcn_wmma_f32_16x16x4_f32(
                false, aH, false, bF, (short)0, p[j], false, false);
        }
    }

    // ---- Epilogue: out[b,k,n0+M,col] = x[...] + P[M][col] + b2[col] ----
    const size_t rowBase = ((size_t)bk * N_ + n0) * D_;
#pragma unroll
    for (int j = 0; j < 4; ++j) {
        const int col = j * 16 + lanelo;
        const float bias2 = b2[col];
#pragma unroll
        for (int v = 0; v < 8; ++v) {
            const int row = v + 8 * hi;
            const size_t off = rowBase + (size_t)row * D_ + col;
            out[off] = x[off] + p[j][v] + bias2;
        }
    }
}

// ---------------------------------------------------------------------------
extern "C" void kernel_launch(void* const* d_in, const int* in_sizes, int n_in,
                              void* d_out, int out_size, void* d_ws, size_t ws_size,
                              hipStream_t stream) {
    const float* xyz = (const float*)d_in[0];
    const float* x   = (const float*)d_in[1];
    const float* W1  = (const float*)d_in[2];
    const float* b1  = (const float*)d_in[3];
    const float* W2  = (const float*)d_in[4];
    const float* b2  = (const float*)d_in[5];
    float* out = (float*)d_out;

    float* delta_ws = (float*)d_ws;   // B*K*N*3 floats = ~6.3 MB

    // Phase 1: KNN + deltas
    knn_delta_kernel<<<(B_ * N_) / 256, 256, 0, stream>>>(xyz, delta_ws);

    // Phase 2: MLP (WMMA f32) + bias + ReLU + bias + x-add
    const int tiles = B_ * K_ * (N_ / 16);   // 32768 tiles, 8 per block
    mlp_add_kernel<<<tiles / 8, 256, 0, stream>>>(delta_ws, x, W1, b1, W2, b2, out);
}